// PAMPJPELoss_12154757448021
// MI455X (gfx1250) — compile-verified
//
#include <hip/hip_runtime.h>

typedef __attribute__((ext_vector_type(2))) float v2f;
typedef __attribute__((ext_vector_type(8))) float v8f;

#define KP   133       // keypoints per sample
#define ROWS 399       // floats per sample matrix (133*3)
#define SPW  5         // samples per wave in phase A (5*3 = 15 rows of the 16x16 tile)

// ---------------------------------------------------------------------------
// Phase A: per-sample raw cross-covariance / sums / Gram via fp32 WMMA.
// Wave handles 5 samples. A-operand rows m = 3s+i map to target coord i of
// sample s; B-operand cols n = 3s+j map to output coord j. 32-bit A layout:
// lane half h holds K = 2h (VGPR0), 2h+1 (VGPR1) -- identical per-lane element
// set as the B layout, so one loaded v2f serves as A or B.
// ---------------------------------------------------------------------------
__global__ void __launch_bounds__(256)
pa_crosscov_wmma(const float* __restrict__ out_pts,   // d_in[0]: output (S2)
                 const float* __restrict__ tgt_pts,   // d_in[1]: target (S1)
                 float* __restrict__ ws, int N)
{
    const int  lane = threadIdx.x & 31;
    const long wave = (long)((blockIdx.x * blockDim.x + threadIdx.x) >> 5);
    const int  half = lane >> 4;
    const int  q    = lane & 15;
    const int  sloc = q / 3;             // 0..4 (q==15 -> 5, idle lane)
    const int  coord= q - 3 * sloc;      // 0..2
    const long sbase = wave * SPW;

    const bool lane_ok = (q < 15) && (sbase + sloc < N);
    long samp = sbase + (sloc < SPW ? sloc : SPW - 1);
    if (samp >= N) samp = N - 1;

    const float* trow = tgt_pts + samp * ROWS + coord;
    const float* orow = out_pts + samp * ROWS + coord;

    v8f accK  = {0.f,0.f,0.f,0.f,0.f,0.f,0.f,0.f};   // raw  sum_k t_i * o_j
    v8f accST = {0.f,0.f,0.f,0.f,0.f,0.f,0.f,0.f};   // sum_k t_i
    v8f accSO = {0.f,0.f,0.f,0.f,0.f,0.f,0.f,0.f};   // sum_k o_i
    v8f accGT = {0.f,0.f,0.f,0.f,0.f,0.f,0.f,0.f};   // sum_k t_i * t_j (Gram)
    const v2f ones = {1.f, 1.f};

    for (int k0 = 0; k0 < KP; k0 += 4) {
        v2f tA, oB;
        #pragma unroll
        for (int r = 0; r < 2; ++r) {
            int  k  = k0 + 2 * half + r;
            bool ok = lane_ok && (k < KP);
            int  kc = (k < KP) ? k : (KP - 1);
            float tv = trow[kc * 3];
            float ov = orow[kc * 3];
            tA[r] = ok ? tv : 0.f;
            oB[r] = ok ? ov : 0.f;
        }
        accK  = __builtin_amdgcn_wmma_f32_16x16x4_f32(false, tA, false, oB,   (short)0, accK,  false, false);
        accST = __builtin_amdgcn_wmma_f32_16x16x4_f32(false, tA, false, ones, (short)0, accST, false, false);
        accSO = __builtin_amdgcn_wmma_f32_16x16x4_f32(false, oB, false, ones, (short)0, accSO, false, false);
        accGT = __builtin_amdgcn_wmma_f32_16x16x4_f32(false, tA, false, tA,   (short)0, accGT, false, false);
    }

    // Extraction: C/D layout -> VGPR r of lane holds (M = r + 8*half, N = lane&15).
    // Keep diagonal 3x3 sample blocks only. SoA stores for coalesced phase B.
    const long Nl  = N;
    float* wsK = ws;              // 9*N
    float* wsT = ws + 9  * Nl;    // 3*N
    float* wsO = ws + 12 * Nl;    // 3*N
    float* wsG = ws + 15 * Nl;    // 3*N
    #pragma unroll
    for (int r = 0; r < 8; ++r) {
        int M  = r + 8 * half;
        int sA = M / 3, iA = M - 3 * sA;
        int sB = sloc,  jB = coord;
        if (sA == sB && sA < SPW) {
            long sg = sbase + sA;
            if (sg < N) {
                wsK[(long)(iA * 3 + jB) * Nl + sg] = accK[r];
                if (jB == 0) { wsT[(long)iA * Nl + sg] = accST[r];
                               wsO[(long)iA * Nl + sg] = accSO[r]; }
                if (iA == jB)  wsG[(long)iA * Nl + sg] = accGT[r];
            }
        }
    }
}

// ---------------------------------------------------------------------------
// Phase B: one lane per sample. Center stats, 3x3 Jacobi SVD, Kabsch R/scale/t.
// ---------------------------------------------------------------------------
#define JROT(APP,AQQ,APQ,ARP,ARQ,VP0,VQ0,VP1,VQ1,VP2,VQ2) do {                \
    float apq_ = APQ;                                                         \
    if (__builtin_fabsf(apq_) > 1e-30f) {                                     \
        float tau_ = (AQQ - APP) / (2.f * apq_);                              \
        float tt_  = (tau_ >= 0.f ? 1.f : -1.f) /                             \
                     (__builtin_fabsf(tau_) + __builtin_sqrtf(1.f + tau_*tau_)); \
        float cc_  = 1.f / __builtin_sqrtf(1.f + tt_*tt_);                    \
        float ss_  = tt_ * cc_;                                               \
        APP -= tt_ * apq_; AQQ += tt_ * apq_; APQ = 0.f;                      \
        float rp_ = ARP, rq_ = ARQ;                                           \
        ARP = cc_*rp_ - ss_*rq_;  ARQ = ss_*rp_ + cc_*rq_;                    \
        float x_;                                                             \
        x_ = VP0; VP0 = cc_*x_ - ss_*VQ0; VQ0 = ss_*x_ + cc_*VQ0;             \
        x_ = VP1; VP1 = cc_*x_ - ss_*VQ1; VQ1 = ss_*x_ + cc_*VQ1;             \
        x_ = VP2; VP2 = cc_*x_ - ss_*VQ2; VQ2 = ss_*x_ + cc_*VQ2;             \
    } } while (0)

#define CSWAP(LA,LB,VA0,VB0,VA1,VB1,VA2,VB2) do {                             \
    if (LA < LB) { float t_;                                                  \
        t_=LA; LA=LB; LB=t_;   t_=VA0; VA0=VB0; VB0=t_;                       \
        t_=VA1; VA1=VB1; VB1=t_; t_=VA2; VA2=VB2; VB2=t_; } } while (0)

__global__ void __launch_bounds__(256)
pa_svd_kabsch(float* __restrict__ ws, int N)
{
    int s = blockIdx.x * blockDim.x + threadIdx.x;
    if (s >= N) return;
    const long Nl = N;
    const float* wsK = ws;
    const float* wsT = ws + 9  * Nl;
    const float* wsO = ws + 12 * Nl;
    const float* wsG = ws + 15 * Nl;
    float* wsR  = ws + 18 * Nl;   // 9*N
    float* wsS  = ws + 27 * Nl;   // 1*N
    float* wsTr = ws + 28 * Nl;   // 3*N

    const float n = (float)KP;
    float m10 = wsT[0*Nl+s]/n, m11 = wsT[1*Nl+s]/n, m12 = wsT[2*Nl+s]/n;  // mu1 (target)
    float m20 = wsO[0*Nl+s]/n, m21 = wsO[1*Nl+s]/n, m22 = wsO[2*Nl+s]/n;  // mu2 (output)

    // centered cross-covariance Kc = X1^T X2 = Kraw - n * mu1 mu2^T
    float k00 = wsK[0*Nl+s] - n*m10*m20,  k01 = wsK[1*Nl+s] - n*m10*m21,  k02 = wsK[2*Nl+s] - n*m10*m22;
    float k10 = wsK[3*Nl+s] - n*m11*m20,  k11 = wsK[4*Nl+s] - n*m11*m21,  k12 = wsK[5*Nl+s] - n*m11*m22;
    float k20 = wsK[6*Nl+s] - n*m12*m20,  k21 = wsK[7*Nl+s] - n*m12*m21,  k22 = wsK[8*Nl+s] - n*m12*m22;

    float var1 = (wsG[0*Nl+s] + wsG[1*Nl+s] + wsG[2*Nl+s]) - n*(m10*m10 + m11*m11 + m12*m12);
    var1 = fmaxf(var1, 1e-20f);

    // A = Kc^T Kc (symmetric PSD)
    float a00 = k00*k00 + k10*k10 + k20*k20;
    float a01 = k00*k01 + k10*k11 + k20*k21;
    float a02 = k00*k02 + k10*k12 + k20*k22;
    float a11 = k01*k01 + k11*k11 + k21*k21;
    float a12 = k01*k02 + k11*k12 + k21*k22;
    float a22 = k02*k02 + k12*k12 + k22*k22;

    float v00=1.f,v01=0.f,v02=0.f, v10=0.f,v11=1.f,v12=0.f, v20=0.f,v21=0.f,v22=1.f;
    #pragma unroll
    for (int sweep = 0; sweep < 6; ++sweep) {
        JROT(a00,a11,a01, a02,a12,  v00,v01, v10,v11, v20,v21);
        JROT(a00,a22,a02, a01,a12,  v00,v02, v10,v12, v20,v22);
        JROT(a11,a22,a12, a01,a02,  v01,v02, v11,v12, v21,v22);
    }

    float l0=a00, l1=a11, l2=a22;           // eigenvalues, columns of V = eigenvectors
    CSWAP(l0,l1, v00,v01, v10,v11, v20,v21);
    CSWAP(l0,l2, v00,v02, v10,v12, v20,v22);
    CSWAP(l1,l2, v01,v02, v11,v12, v21,v22);

    // v2 = v0 x v1 (forces det(V)=+1)
    float w0 = v10*v21 - v20*v11;
    float w1 = v20*v01 - v00*v21;
    float w2 = v00*v11 - v10*v01;

    // u0 = normalize(Kc v0), s0 = |Kc v0|
    float c0x = k00*v00 + k01*v10 + k02*v20;
    float c0y = k10*v00 + k11*v10 + k12*v20;
    float c0z = k20*v00 + k21*v10 + k22*v20;
    float s0  = __builtin_sqrtf(c0x*c0x + c0y*c0y + c0z*c0z);
    float i0  = 1.f / fmaxf(s0, 1e-20f);
    float u0x = c0x*i0, u0y = c0y*i0, u0z = c0z*i0;

    float c1x = k00*v01 + k01*v11 + k02*v21;
    float c1y = k10*v01 + k11*v11 + k12*v21;
    float c1z = k20*v01 + k21*v11 + k22*v21;
    float s1  = __builtin_sqrtf(c1x*c1x + c1y*c1y + c1z*c1z);
    float i1  = 1.f / fmaxf(s1, 1e-20f);
    float u1x = c1x*i1, u1y = c1y*i1, u1z = c1z*i1;
    float d01 = u1x*u0x + u1y*u0y + u1z*u0z;     // orthogonality cleanup
    u1x -= d01*u0x; u1y -= d01*u0y; u1z -= d01*u0z;
    float ln = __builtin_sqrtf(u1x*u1x + u1y*u1y + u1z*u1z);
    float il = 1.f / fmaxf(ln, 1e-20f);
    u1x *= il; u1y *= il; u1z *= il;

    // u2 = u0 x u1 (det(U)=+1); signed third singular value gives reflection fix
    float u2x = u0y*u1z - u0z*u1y;
    float u2y = u0z*u1x - u0x*u1z;
    float u2z = u0x*u1y - u0y*u1x;
    float c2x = k00*w0 + k01*w1 + k02*w2;
    float c2y = k10*w0 + k11*w1 + k12*w2;
    float c2z = k20*w0 + k21*w1 + k22*w2;
    float s2s = u2x*c2x + u2y*c2y + u2z*c2z;     // = detsign * s2

    float scale = (s0 + s1 + s2s) / var1;        // trace(R K)/var1

    // R = V U^T (with the det-+1 constructions this equals Vh^T Z U^T)
    float r00 = v00*u0x + v01*u1x + w0*u2x;
    float r01 = v00*u0y + v01*u1y + w0*u2y;
    float r02 = v00*u0z + v01*u1z + w0*u2z;
    float r10 = v10*u0x + v11*u1x + w1*u2x;
    float r11 = v10*u0y + v11*u1y + w1*u2y;
    float r12 = v10*u0z + v11*u1z + w1*u2z;
    float r20 = v20*u0x + v21*u1x + w2*u2x;
    float r21 = v20*u0y + v21*u1y + w2*u2y;
    float r22 = v20*u0z + v21*u1z + w2*u2z;

    float t0 = m20 - scale*(r00*m10 + r01*m11 + r02*m12);
    float t1 = m21 - scale*(r10*m10 + r11*m11 + r12*m12);
    float t2 = m22 - scale*(r20*m10 + r21*m11 + r22*m12);

    wsR[0*Nl+s]=r00; wsR[1*Nl+s]=r01; wsR[2*Nl+s]=r02;
    wsR[3*Nl+s]=r10; wsR[4*Nl+s]=r11; wsR[5*Nl+s]=r12;
    wsR[6*Nl+s]=r20; wsR[7*Nl+s]=r21; wsR[8*Nl+s]=r22;
    wsS[s] = scale;
    wsTr[0*Nl+s]=t0; wsTr[1*Nl+s]=t1; wsTr[2*Nl+s]=t2;
}

// ---------------------------------------------------------------------------
// Phase C: wave per sample; apply transform, per-joint norms, wave reduce,
// write deterministic per-sample partial sum.
// ---------------------------------------------------------------------------
__global__ void __launch_bounds__(256)
pa_dist(const float* __restrict__ out_pts, const float* __restrict__ tgt_pts,
        const float* __restrict__ ws, float* __restrict__ part, int N)
{
    const int  lane = threadIdx.x & 31;
    const long s    = (long)((blockIdx.x * blockDim.x + threadIdx.x) >> 5);
    if (s >= N) return;
    const long Nl = N;
    const float* wsR  = ws + 18 * Nl;
    const float* wsS  = ws + 27 * Nl;
    const float* wsTr = ws + 28 * Nl;

    float r00=wsR[0*Nl+s], r01=wsR[1*Nl+s], r02=wsR[2*Nl+s];
    float r10=wsR[3*Nl+s], r11=wsR[4*Nl+s], r12=wsR[5*Nl+s];
    float r20=wsR[6*Nl+s], r21=wsR[7*Nl+s], r22=wsR[8*Nl+s];
    float sc = wsS[s];
    float t0=wsTr[0*Nl+s], t1=wsTr[1*Nl+s], t2=wsTr[2*Nl+s];

    const float* tb = tgt_pts + s * ROWS;
    const float* ob = out_pts + s * ROWS;
    float acc = 0.f;
    #pragma unroll
    for (int it = 0; it < 5; ++it) {
        int j = lane + 32 * it;
        if (j < KP) {
            float px = tb[j*3+0], py = tb[j*3+1], pz = tb[j*3+2];
            float qx = ob[j*3+0], qy = ob[j*3+1], qz = ob[j*3+2];
            float yx = sc*(r00*px + r01*py + r02*pz) + t0;
            float yy = sc*(r10*px + r11*py + r12*pz) + t1;
            float yz = sc*(r20*px + r21*py + r22*pz) + t2;
            float dx = qx - yx, dy = qy - yy, dz = qz - yz;
            acc += __builtin_sqrtf(dx*dx + dy*dy + dz*dz);
        }
    }
    #pragma unroll
    for (int off = 16; off > 0; off >>= 1)
        acc += __shfl_xor(acc, off, 32);
    if (lane == 0) part[s] = acc;
}

// ---------------------------------------------------------------------------
// Finalize: deterministic single-block tree reduction, mean over N*KP.
// ---------------------------------------------------------------------------
__global__ void __launch_bounds__(256)
pa_finalize(const float* __restrict__ part, float* __restrict__ out, int N)
{
    __shared__ float red[256];
    float a = 0.f;
    for (int i = threadIdx.x; i < N; i += 256) a += part[i];
    red[threadIdx.x] = a;
    __syncthreads();
    #pragma unroll
    for (int st = 128; st > 0; st >>= 1) {
        if ((int)threadIdx.x < st) red[threadIdx.x] += red[threadIdx.x + st];
        __syncthreads();
    }
    if (threadIdx.x == 0) out[0] = red[0] / ((float)N * (float)KP);
}

extern "C" void kernel_launch(void* const* d_in, const int* in_sizes, int n_in,
                              void* d_out, int out_size, void* d_ws, size_t ws_size,
                              hipStream_t stream) {
    const float* outp = (const float*)d_in[0];   // "output"
    const float* tgtp = (const float*)d_in[1];   // "target"
    float* o  = (float*)d_out;
    float* ws = (float*)d_ws;
    const int N = in_sizes[0] / ROWS;            // 65536
    float* part = ws + (long)31 * N;             // N partial sums (total ws: 32*N floats)

    const int wavesA = (N + SPW - 1) / SPW;
    const int thrA   = wavesA * 32;
    pa_crosscov_wmma<<<(thrA + 255) / 256, 256, 0, stream>>>(outp, tgtp, ws, N);
    pa_svd_kabsch  <<<(N + 255) / 256,     256, 0, stream>>>(ws, N);
    pa_dist        <<<((long)N * 32 + 255) / 256, 256, 0, stream>>>(outp, tgtp, ws, part, N);
    pa_finalize    <<<1, 256, 0, stream>>>(part, o, N);
}